// recovery_loss_78271484003258
// MI455X (gfx1250) — compile-verified
//
#include <hip/hip_runtime.h>

#define B_ 128
#define N_ 4096
#define P_ 4096
#define K_ 100
#define KP 112            // 7 * 16 padded candidates
#define INTMAX_ 0x7fffffff

typedef __attribute__((ext_vector_type(2))) float v2f;
typedef __attribute__((ext_vector_type(8))) float v8f;

// ---------------- Kernel A: per-row shift + gather of X100 / dvals ----------------
__global__ __launch_bounds__(256) void prep_kernel(
    const float* __restrict__ preds, const float* __restrict__ target,
    const float* __restrict__ Xffp,  const float* __restrict__ amp,
    float* __restrict__ ysw, float* __restrict__ dvw)
{
  const int b = blockIdx.x;
  const int tid = threadIdx.x;
  const float* trow = target + (size_t)b * N_;
  const float* prow = preds  + (size_t)b * N_;
  const float* xrow = Xffp   + (size_t)b * N_;
  const float* arow = amp    + (size_t)b * N_;

  __shared__ int s_jleft;   // min index with u >= 0
  __shared__ int s_imax;    // max index with u <= 0
  __shared__ int s_shift;
  if (tid == 0) { s_jleft = INTMAX_; s_imax = -1; }
  __syncthreads();

  int ljl = INTMAX_, lim = -1;
  for (int i = tid; i < N_; i += 256) {
    float u = trow[i] / arow[i];
    if (u >= 0.0f && i < ljl) ljl = i;
    if (u <= 0.0f && i > lim) lim = i;
  }
  if (ljl != INTMAX_) atomicMin(&s_jleft, ljl);
  if (lim >= 0)       atomicMax(&s_imax, lim);
  __syncthreads();

  if (tid == 0) {
    float first = trow[0]      / arow[0];
    float last  = trow[N_ - 1] / arow[N_ - 1];
    int j_left  = (s_jleft == INTMAX_) ? 0 : s_jleft;          // argmax of all-False -> 0
    int m_right = (s_imax < 0) ? 0 : (N_ - 1 - s_imax);
    bool left_case = (last < 0.0f) && (first < 0.0f);
    s_shift = left_case ? -(j_left + 1) : m_right;
  }
  __syncthreads();

  const int shift = s_shift;
  if (tid < KP) {
    float y, d;
    if (tid < K_) {
      int src = ((tid - shift) % N_ + N_) % N_;                // python-style mod
      y = xrow[src];
      d = (trow[src] - prow[src]) / arow[src];                 // nu_s - nui_s
    } else { y = 1.0e6f; d = 0.0f; }                           // pad: never wins argmin
    ysw[b * KP + tid] = y;
    dvw[b * KP + tid] = d;
  }
}

// ---------------- Kernel B: WMMA-scored 1-NN + per-batch variance ----------------
// score(k,p) = -2*x_p*y_k + y_k^2  ==  (x_p - y_k)^2 - x_p^2  (monotone in |x_p - y_k| per p)
__global__ __launch_bounds__(256) void nn_var_kernel(
    const float* __restrict__ pointx,
    const float* __restrict__ ysw, const float* __restrict__ dvw,
    float* __restrict__ losses)
{
  const int b    = blockIdx.x;
  const int tid  = threadIdx.x;
  const int lane = tid & 31;
  const int wave = tid >> 5;
  const int half = lane >> 4;          // 0: K-slots 0/1 of A,B ; 1: K-slots 2/3 (zeros)
  const int pl   = lane & 15;

  __shared__ float s_ys[KP];
  __shared__ float s_dv[KP];
  if (tid < KP) { s_ys[tid] = ysw[b * KP + tid]; s_dv[tid] = dvw[b * KP + tid]; }
  __syncthreads();

  double sum = 0.0, sumsq = 0.0;

  // 256 p-tiles of 16; 8 waves * 32 tiles each
  for (int t = wave * 32; t < wave * 32 + 32; ++t) {
    const int p = t * 16 + pl;
    // B operand: col n=pl holds (x_p, 1); lanes 16-31 supply K=2,3 rows = 0
    v2f bop;
    bop.x = (half == 0) ? pointx[p] : 0.0f;
    bop.y = (half == 0) ? 1.0f      : 0.0f;

    float best = 3.0e38f;
    int   bk   = 0;
    for (int kt = 0; kt < 7; ++kt) {
      // A operand: row m=pl holds (-2*y, y^2); lanes 16-31 supply K=2,3 cols = 0
      float y = (half == 0) ? s_ys[kt * 16 + pl] : 0.0f;
      v2f a; a.x = -2.0f * y; a.y = y * y;
      v8f c = {0.f,0.f,0.f,0.f,0.f,0.f,0.f,0.f};
      v8f d = __builtin_amdgcn_wmma_f32_16x16x4_f32(
          /*neg_a=*/false, a, /*neg_b=*/false, bop,
          /*c_mod=*/(short)0, c, /*reuse_a=*/false, /*reuse_b=*/false);
      // lane<16: D(M=r, N=pl) => k = kt*16+r ; lane>=16: D(M=r+8, N=pl) => k = kt*16+8+r
      const int kbase = kt * 16 + (half ? 8 : 0);
#pragma unroll
      for (int r = 0; r < 8; ++r) {
        float s = d[r];
        bool take = (s < best);               // strict '<' keeps first (lowest) index
        bk   = take ? (kbase + r) : bk;
        best = take ? s : best;
      }
    }
    // merge candidate halves held by lane pair (l, l+16) — same p
    float ob = __shfl_xor(best, 16, 32);
    int   ok = __shfl_xor(bk,   16, 32);
    bool take = (ob < best) || (ob == best && ok < bk);
    best = take ? ob : best;
    bk   = take ? ok : bk;

    float dv = s_dv[bk];
    // only half 0 contributes (halves are duplicates); skip p==0 per diff[:,1:]
    double contrib = (half == 0 && p != 0) ? (double)dv : 0.0;
    sum   += contrib;
    sumsq += contrib * contrib;
  }

  __shared__ double rs[256];
  __shared__ double rq[256];
  rs[tid] = sum; rq[tid] = sumsq;
  __syncthreads();
  for (int off = 128; off > 0; off >>= 1) {
    if (tid < off) { rs[tid] += rs[tid + off]; rq[tid] += rq[tid + off]; }
    __syncthreads();
  }
  if (tid == 0) {
    const double M = (double)(P_ - 1);           // 4095 samples, population variance
    double mean = rs[0] / M;
    losses[b] = (float)(rq[0] / M - mean * mean);
  }
}

// ---------------- Kernel C: mean over batches ----------------
__global__ __launch_bounds__(128) void finalize_kernel(
    const float* __restrict__ losses, float* __restrict__ out)
{
  const int tid = threadIdx.x;
  __shared__ float r[128];
  r[tid] = losses[tid];
  __syncthreads();
  for (int off = 64; off > 0; off >>= 1) {
    if (tid < off) r[tid] += r[tid + off];
    __syncthreads();
  }
  if (tid == 0) out[0] = r[0] / (float)B_;
}

extern "C" void kernel_launch(void* const* d_in, const int* in_sizes, int n_in,
                              void* d_out, int out_size, void* d_ws, size_t ws_size,
                              hipStream_t stream)
{
  const float* preds  = (const float*)d_in[0];
  const float* target = (const float*)d_in[1];
  const float* Xffp   = (const float*)d_in[2];
  const float* amp    = (const float*)d_in[3];
  const float* pointx = (const float*)d_in[4];

  float* ysw    = (float*)d_ws;            // [128][112]
  float* dvw    = ysw + B_ * KP;           // [128][112]
  float* losses = dvw + B_ * KP;           // [128]

  prep_kernel<<<B_, 256, 0, stream>>>(preds, target, Xffp, amp, ysw, dvw);
  nn_var_kernel<<<B_, 256, 0, stream>>>(pointx, ysw, dvw, losses);
  finalize_kernel<<<1, 128, 0, stream>>>(losses, (float*)d_out);
}